// RegionLoss_90958817394780
// MI455X (gfx1250) — compile-verified
//
#include <hip/hip_runtime.h>
#include <hip/hip_bf16.h>

#define NB 64
#define NA 5
#define NC 20
#define NH 64
#define NW 64
#define MAXB 50
#define HW (NH * NW)                 // 4096
#define CELLS_PER_B (NA * HW)        // 20480
#define NTOT (NB * CELLS_PER_B)      // 1310720
#define RECF 16                      // floats per box record

__constant__ float c_AW[NA] = {1.3221f, 3.19275f, 5.05587f, 9.47112f, 11.2364f};
__constant__ float c_AH[NA] = {1.73145f, 4.00944f, 8.09892f, 4.84053f, 10.0071f};

// Box record layout (16 floats):
// 0 xlo, 1 xhi, 2 ylo, 3 yhi, 4 gw, 5 gh, 6 garea,
// 7 tx, 8 ty, 9 tw, 10 th, 11 cellcode(int), 12 tcls(int), 13-15 pad

__global__ void region_zero(float* loss) { *loss = 0.0f; }

__global__ void region_phase1(const float* __restrict__ target,
                              float* __restrict__ box) {
    const int b = blockIdx.x;
    const int t = threadIdx.x;
    __shared__ float sbx[MAXB];
    const float* tg = target + b * MAXB * 5;
    if (t < MAXB) sbx[t] = tg[t * 5 + 1];
    __syncthreads();
    if (t >= MAXB) return;

    // prefix validity: loop breaks at first x==0
    bool valid = true;
    for (int s = 0; s <= t; ++s) valid = valid && (sbx[s] != 0.0f);

    const float cls = tg[t * 5 + 0];
    const float gx = tg[t * 5 + 1] * (float)NW;
    const float gy = tg[t * 5 + 2] * (float)NH;
    const float gw = tg[t * 5 + 3] * (float)NW;
    const float gh = tg[t * 5 + 4] * (float)NH;
    const int gi = (int)gx;
    const int gj = (int)gy;

    // best anchor by shape-only IoU at origin: carea = min(aw,gw)*min(ah,gh)
    float best = -1.0f;
    int bn = 0;
    #pragma unroll
    for (int n = 0; n < NA; ++n) {
        const float aw = c_AW[n], ah = c_AH[n];
        const float ca = fminf(aw, gw) * fminf(ah, gh);
        const float iou = ca / (aw * ah + gw * gh - ca);
        if (iou > best) { best = iou; bn = n; }
    }

    float* r = box + (size_t)(b * MAXB + t) * RECF;
    int* ri = (int*)r;
    if (valid) {
        r[0] = gx - 0.5f * gw;  r[1] = gx + 0.5f * gw;
        r[2] = gy - 0.5f * gh;  r[3] = gy + 0.5f * gh;
        r[4] = gw;  r[5] = gh;  r[6] = gw * gh;
        r[7] = gx - (float)gi;  r[8] = gy - (float)gj;
        r[9] = gw / c_AW[bn];   r[10] = gh / c_AH[bn];
        ri[11] = (bn * NH + gj) * NW + gi;
        ri[12] = (int)cls;
    } else {
        // forces carea==0, uarea==parea>0 -> iou==0; never matches any cell
        r[0] =  1e30f;  r[1] = -1e30f;
        r[2] =  1e30f;  r[3] = -1e30f;
        r[4] = 0.0f; r[5] = 0.0f; r[6] = 0.0f;
        r[7] = 0.0f; r[8] = 0.0f; r[9] = 0.0f; r[10] = 0.0f;
        ri[11] = -1;
        ri[12] = 0;
    }
    r[13] = 0.0f; r[14] = 0.0f; r[15] = 0.0f;
}

__global__ __launch_bounds__(256) void region_phase2(
    const float* __restrict__ out, const float* __restrict__ box,
    float* __restrict__ loss) {
    __shared__ float sb[MAXB * RECF];
    __shared__ float wsum[8];

    const int idx = blockIdx.x * 256 + threadIdx.x;
    const int b = idx / CELLS_PER_B;   // uniform per block (80 blocks/batch)

    // cooperative load of this batch's 50-box table into LDS
    const float* src = box + (size_t)b * MAXB * RECF;
    for (int k = threadIdx.x; k < MAXB * RECF; k += 256) sb[k] = src[k];
    __syncthreads();

    const int rem  = idx % CELLS_PER_B;
    const int a    = rem / HW;         // uniform per block
    const int cell = rem % HW;
    const int j = cell / NW, i = cell % NW;
    const float aw = c_AW[a], ah = c_AH[a];

    const float* p = out + (size_t)(b * (NA * 25) + a * 25) * HW + cell;
    const float f0 = p[0];
    const float f1 = p[HW];
    const float f2 = p[2 * HW];
    const float f3 = p[3 * HW];
    const float f4 = p[4 * HW];

    const float sx   = 1.0f / (1.0f + __expf(-f0));
    const float sy   = 1.0f / (1.0f + __expf(-f1));
    const float ew   = __expf(f2);
    const float eh   = __expf(f3);
    const float conf = 1.0f / (1.0f + __expf(-f4));

    const float px = sx + (float)i, py = sy + (float)j;
    const float pw = ew * aw, ph = eh * ah;
    const float pxh = px + 0.5f * pw, pxl = px - 0.5f * pw;
    const float pyh = py + 0.5f * ph, pyl = py - 0.5f * ph;
    const float parea = pw * ph;
    const int mycode = (a * NH + j) * NW + i;

    bool silenced = false, is_t = false;
    float ttx = 0.f, tty = 0.f, ttw = 0.f, tth = 0.f;
    float tca = 0.f, tua = 1.f;
    int tcls = 0;

    // 50-box loop: division-free silencing test (iou>0.6 <=> ca > 0.6*ua)
    #pragma unroll 5
    for (int t = 0; t < MAXB; ++t) {
        const float* r = sb + t * RECF;
        const float xlo = r[0], xhi = r[1], ylo = r[2], yhi = r[3];
        const float gw = r[4], gh = r[5], ga = r[6];
        const float Mx = fmaxf(pxh, xhi), mx = fminf(pxl, xlo);
        const float My = fmaxf(pyh, yhi), my = fminf(pyl, ylo);
        const float cw = pw + gw - (Mx - mx);
        const float ch = ph + gh - (My - my);
        const float ca = (fminf(cw, ch) <= 0.0f) ? 0.0f : cw * ch;
        const float ua = parea + ga - ca;
        silenced = silenced || (ca > 0.6f * ua);
        const int code = ((const int*)r)[11];
        if (code == mycode) {   // last match wins (sequential-scatter semantics)
            is_t = true;
            ttx = r[7]; tty = r[8]; ttw = r[9]; tth = r[10];
            tcls = ((const int*)r)[12];
            tca = ca; tua = ua;
        }
    }

    float acc;
    if (is_t) {
        const float tconf = tca / tua;                 // only here we divide
        const float dx = sx - ttx, dy = sy - tty;
        const float dw = ew - ttw, dh = eh - tth;      // loss_w/h use exp only
        const float dc = conf - tconf;                 // conf_mask = OBJ = 1
        acc = 0.5f * (dx * dx + dy * dy + dw * dw + dh * dh + dc * dc);
        // class cross-entropy at the assigned cell
        const float* cp = p + 5 * HW;
        float lg[NC], m = -1e30f;
        #pragma unroll
        for (int c = 0; c < NC; ++c) { lg[c] = cp[c * HW]; m = fmaxf(m, lg[c]); }
        float se = 0.0f;
        #pragma unroll
        for (int c = 0; c < NC; ++c) se += __expf(lg[c] - m);
        acc += __logf(se) + m - lg[tcls];
    } else {
        // sqrt(conf_mask) is 0 or 1; tconf == 0 at non-target cells
        acc = silenced ? 0.0f : 0.5f * conf * conf;
    }

    // ---- wave reduction via V_WMMA_F32_16X16X4_F32 (B = ones) ----
    // A[16x4]: lanes 0-15 supply K=0 (vgpr0), lanes 16-31 supply K=2.
    // D[m][n] = acc[m] + acc[m+16]; lane n<16 sums rows 0-7, lanes>=16 rows 8-15.
    float s;
#if defined(__gfx1250__) && __has_builtin(__builtin_amdgcn_wmma_f32_16x16x4_f32)
    {
        typedef float v2f __attribute__((ext_vector_type(2)));
        typedef float v8f __attribute__((ext_vector_type(8)));
        v2f A; A[0] = acc; A[1] = 0.0f;
        v2f Bm; Bm[0] = 1.0f; Bm[1] = 1.0f;
        v8f C = {};
        v8f D = __builtin_amdgcn_wmma_f32_16x16x4_f32(
            false, A, false, Bm, (short)0, C, false, false);
        s = D[0] + D[1] + D[2] + D[3] + D[4] + D[5] + D[6] + D[7];
        s += __shfl_xor(s, 16, 32);    // combine the two half-row groups
    }
#else
    s = acc;
    for (int o = 16; o > 0; o >>= 1) s += __shfl_xor(s, o, 32);
#endif

    const int lane = threadIdx.x & 31, wv = threadIdx.x >> 5;
    if (lane == 0) wsum[wv] = s;
    __syncthreads();
    if (threadIdx.x == 0) {
        float tot = 0.0f;
        #pragma unroll
        for (int k = 0; k < 8; ++k) tot += wsum[k];
        atomicAdd(loss, tot);
    }
}

extern "C" void kernel_launch(void* const* d_in, const int* in_sizes, int n_in,
                              void* d_out, int out_size, void* d_ws, size_t ws_size,
                              hipStream_t stream) {
    (void)in_sizes; (void)n_in; (void)out_size; (void)ws_size;
    const float* output = (const float*)d_in[0];   // [64,125,64,64] fp32
    const float* target = (const float*)d_in[1];   // [64,250] fp32
    float* loss = (float*)d_out;                   // scalar fp32
    float* box  = (float*)d_ws;                    // 64*50*16 floats = 200 KB

    region_zero<<<1, 1, 0, stream>>>(loss);
    region_phase1<<<NB, 64, 0, stream>>>(target, box);
    region_phase2<<<NTOT / 256, 256, 0, stream>>>(output, box, loss);
}